// GAT_23768349016460
// MI455X (gfx1250) — compile-verified
//
#include <hip/hip_runtime.h>
#include <hip/hip_bf16.h>
#include <math.h>

// ---------------------------------------------------------------------------
// 3-layer GATConv for MI455X (gfx1250, wave32).
//  - Dense projection: fp32 WMMA (v_wmma_f32_16x16x4_f32), exact precision.
//    W staged into LDS via Tensor Data Mover (tensor_load_to_lds + TENSORcnt),
//    padded rows for bank-conflict-free fragment reads.
//  - Segment softmax: 3 streaming edge passes, L2-resident node arrays.
//    max   -> ordered-uint atomicMax   (global_atomic_max_u32)
//    sum   -> hardware f32 atomic add  (global_atomic_add_f32)
//    aggr  -> hardware f32 atomic add
// ---------------------------------------------------------------------------

typedef __attribute__((ext_vector_type(2))) float v2f;
typedef __attribute__((ext_vector_type(8))) float v8f;
typedef unsigned int u32x4 __attribute__((ext_vector_type(4)));
typedef int i32x8 __attribute__((ext_vector_type(8)));
typedef int i32x4 __attribute__((ext_vector_type(4)));

#define NN 100000      // nodes
#define NE 1600000     // edges
#define KDIM 128       // GEMM K for all three layers (128-dim inputs)
#define LDS_PAD 8      // dwords of padding per staged W row

#define ENC_NEG_INF 0x007FFFFFu   // ordEnc(-inf)

__device__ __forceinline__ unsigned ordEnc(float x) {
  unsigned u = __float_as_uint(x);
  return (u & 0x80000000u) ? ~u : (u | 0x80000000u);
}
__device__ __forceinline__ float ordDec(unsigned u) {
  unsigned v = (u & 0x80000000u) ? (u & 0x7FFFFFFFu) : ~u;
  return __uint_as_float(v);
}
__device__ __forceinline__ void atomAddF(float* p, float v) {
  __hip_atomic_fetch_add(p, v, __ATOMIC_RELAXED, __HIP_MEMORY_SCOPE_AGENT);
}

// ---------------------------------------------------------------------------
// C[M,Nc] = A[M,128] * B[128,Nc], fp32 WMMA 16x16x4.
// One wave per 16-row M-tile; A fragments (32 x v2f = 64 VGPRs) preloaded once
// and reused across all Nc/16 N-tiles.  B (= W, <=64KB) is staged to LDS by
// the Tensor Data Mover with 8-dword row padding: the two rows a B-fragment
// needs are then 16 LDS banks apart -> conflict-free half-wave reads.
// ---------------------------------------------------------------------------
__global__ void __launch_bounds__(128)
gat_gemm_wmma(const float* __restrict__ A, const float* __restrict__ B,
              float* __restrict__ C, int M, int Nc)
{
  extern __shared__ float Bs[];
  const int SNC  = Nc + LDS_PAD;            // padded LDS row stride (dwords)
  const int wave = threadIdx.x >> 5;
  const int lane = threadIdx.x & 31;

  // ---- Stage W[128,Nc] into LDS ----
#if defined(__gfx1250__) && __has_builtin(__builtin_amdgcn_tensor_load_to_lds)
  if (wave == 0) {                           // one TDM issue per workgroup
    const unsigned lds_base = (unsigned)(uintptr_t)Bs;   // low 32b = LDS offset
    const unsigned long long ga = (unsigned long long)(uintptr_t)B;
    u32x4 g0;
    g0[0] = 1u;                                          // count=1, user mode
    g0[1] = lds_base;                                    // lds_addr
    g0[2] = (unsigned)ga;                                // global_addr[31:0]
    g0[3] = (unsigned)((ga >> 32) & 0x01FFFFFFull) | (2u << 30); // addr hi | type=2
    const int pad_iv = (Nc == 128) ? 6 : 4;  // pad every Nc dwords (2^(v+1))
    i32x8 g1;
    g1[0] = (2 << 16) | (1 << 20) | (pad_iv << 22) | (7 << 25); // 4B elems, pad_en, pad 8dw
    g1[1] = (Nc & 0xFFFF) << 16;             // tensor_dim0[15:0]
    g1[2] = (Nc >> 16) | (128 << 16);        // tensor_dim0 hi | tensor_dim1 lo
    g1[3] = (Nc << 16);                      // tensor_dim1 hi(=0) | tile_dim0
    g1[4] = 128;                             // tile_dim1=128, tile_dim2=0
    g1[5] = Nc;                              // tensor_dim0_stride[31:0]
    g1[6] = 0;
    g1[7] = 0;
    i32x4 gz4 = {0, 0, 0, 0};                // 2D tensor: groups 2/3 unused
    i32x8 gz8 = {0, 0, 0, 0, 0, 0, 0, 0};
    __builtin_amdgcn_tensor_load_to_lds(g0, g1, gz4, gz4, gz8, 0);
    __builtin_amdgcn_s_wait_tensorcnt(0);
  }
  __syncthreads();
#else
  for (int idx = threadIdx.x; idx < 128 * Nc; idx += 128) {
    int row = idx / Nc, col = idx - row * Nc;
    Bs[row * SNC + col] = B[idx];
  }
  __syncthreads();
#endif

  const int mt = blockIdx.x * 4 + wave;      // wave-uniform guard (EXEC all-1s)
  if (mt * 16 < M) {
    const int r  = lane & 15;
    const int hi = lane >> 4;

    const float* Arow = A + (size_t)(mt * 16 + r) * KDIM + 2 * hi;
    v2f afrag[KDIM / 4];
#pragma unroll
    for (int kk = 0; kk < KDIM / 4; ++kk)
      afrag[kk] = *(const v2f*)(Arow + kk * 4);

    const int NT = Nc >> 4;
    for (int nt = 0; nt < NT; ++nt) {
      const float* Bc = Bs + nt * 16 + r;
      v2f bfrag[KDIM / 4];
#pragma unroll
      for (int kk = 0; kk < KDIM / 4; ++kk) {
        const int k0 = kk * 4 + 2 * hi;
        bfrag[kk].x = Bc[k0 * SNC];
        bfrag[kk].y = Bc[(k0 + 1) * SNC];
      }
      v8f acc = {};
#pragma unroll
      for (int kk = 0; kk < KDIM / 4; ++kk)
        acc = __builtin_amdgcn_wmma_f32_16x16x4_f32(false, afrag[kk],
                                                    false, bfrag[kk],
                                                    (short)0, acc, false, false);
      float* Cp = C + (size_t)(mt * 16 + hi * 8) * Nc + nt * 16 + r;
#pragma unroll
      for (int j = 0; j < 8; ++j)
        Cp[(size_t)j * Nc] = acc[j];
    }
  }
}

// ---------------------------------------------------------------------------
// el[n,h] = dot(f[n,h,:], al[h,:]);  er likewise.  D = 32 = one wave.
// ---------------------------------------------------------------------------
__global__ void __launch_bounds__(128)
gat_attn_coef(const float* __restrict__ f, const float* __restrict__ al,
              const float* __restrict__ ar, float* __restrict__ el,
              float* __restrict__ er, int H)
{
  const int wave = threadIdx.x >> 5;
  const int lane = threadIdx.x & 31;
  const int gw = blockIdx.x * 4 + wave;
  if (gw >= NN * H) return;
  const int n = gw / H;
  const int h = gw - n * H;
  const float v = f[(size_t)n * (H * 32) + h * 32 + lane];
  float a = v * al[h * 32 + lane];
  float b = v * ar[h * 32 + lane];
#pragma unroll
  for (int o = 16; o > 0; o >>= 1) {
    a += __shfl_xor(a, o, 32);
    b += __shfl_xor(b, o, 32);
  }
  if (lane == 0) { el[gw] = a; er[gw] = b; }
}

__global__ void gat_init_ms(unsigned* __restrict__ m, float* __restrict__ s, int n) {
  int t = blockIdx.x * blockDim.x + threadIdx.x;
  if (t < n) { m[t] = ENC_NEG_INF; s[t] = 0.0f; }
}

__global__ void gat_init_bias(float* __restrict__ out, const float* __restrict__ b,
                              int total, int HD) {
  int t = blockIdx.x * blockDim.x + threadIdx.x;
  if (t < total) out[t] = b[t % HD];
}

// Pass 1: e = leaky_relu(el[src]+er[dst], 0.2); segment-max via ordered-uint atomicMax.
__global__ void gat_edge_score(const float* __restrict__ el, const float* __restrict__ er,
                               const int* __restrict__ src, const int* __restrict__ dst,
                               float* __restrict__ ew, unsigned* __restrict__ mE, int H)
{
  int t = blockIdx.x * blockDim.x + threadIdx.x;
  if (t >= NE * H) return;
  int e = t / H, h = t - e * H;
  float v = el[src[e] * H + h] + er[dst[e] * H + h];
  v = v > 0.0f ? v : 0.2f * v;
  ew[t] = v;
  atomicMax(&mE[dst[e] * H + h], ordEnc(v));
}

// Pass 2: w = exp(e - m[dst]);  s[dst] += w.
__global__ void gat_edge_exp(const int* __restrict__ dst, const unsigned* __restrict__ mE,
                             float* __restrict__ ew, float* __restrict__ ss, int H)
{
  int t = blockIdx.x * blockDim.x + threadIdx.x;
  if (t >= NE * H) return;
  int e = t / H, h = t - e * H;
  int di = dst[e] * H + h;
  float m = ordDec(mE[di]);
  if (!isfinite(m)) m = 0.0f;               // zero in-degree safety (matches ref)
  float w = __expf(ew[t] - m);
  ew[t] = w;
  atomAddF(&ss[di], w);
}

// Pass 3: out[dst,:] += (w / max(s[dst],1e-9)) * f[src,:].
__global__ void gat_aggregate(const int* __restrict__ src, const int* __restrict__ dst,
                              const float* __restrict__ ew, const float* __restrict__ ss,
                              const float* __restrict__ f, float* __restrict__ out, int H)
{
  const int HD = H * 32;
  int t = blockIdx.x * blockDim.x + threadIdx.x;
  if (t >= NE * HD) return;
  int e = t / HD;
  int j = t - e * HD;
  int h = j >> 5;
  int d = dst[e];
  float coef = ew[e * H + h] / fmaxf(ss[d * H + h], 1e-9f);
  atomAddF(&out[(size_t)d * HD + j], coef * f[(size_t)src[e] * HD + j]);
}

__global__ void gat_elu(float* __restrict__ x, int n) {
  int t = blockIdx.x * blockDim.x + threadIdx.x;
  if (t < n) { float v = x[t]; x[t] = v > 0.0f ? v : expm1f(v); }
}

// ---------------------------------------------------------------------------
static void run_gat_layer(const float* hin, const float* W, const float* al,
                          const float* ar, const float* b,
                          const int* src, const int* dst,
                          float* f, float* out, float* ew, float* el, float* er,
                          unsigned* mE, float* ss, int H, bool elu,
                          hipStream_t stream)
{
  const int HD = H * 32;
  const int mtiles = (NN + 15) / 16;          // 6250
  const size_t ldsBytes = (size_t)128 * (HD + LDS_PAD) * sizeof(float);
  gat_gemm_wmma<<<(mtiles + 3) / 4, 128, ldsBytes, stream>>>(hin, W, f, NN, HD);
  gat_attn_coef<<<(NN * H + 3) / 4, 128, 0, stream>>>(f, al, ar, el, er, H);
  gat_init_ms<<<(NN * H + 255) / 256, 256, 0, stream>>>(mE, ss, NN * H);
  gat_init_bias<<<(NN * HD + 255) / 256, 256, 0, stream>>>(out, b, NN * HD, HD);
  gat_edge_score<<<(NE * H + 255) / 256, 256, 0, stream>>>(el, er, src, dst, ew, mE, H);
  gat_edge_exp<<<(NE * H + 255) / 256, 256, 0, stream>>>(dst, mE, ew, ss, H);
  gat_aggregate<<<(NE * HD + 255) / 256, 256, 0, stream>>>(src, dst, ew, ss, f, out, H);
  if (elu)
    gat_elu<<<(NN * HD + 255) / 256, 256, 0, stream>>>(out, NN * HD);
}

extern "C" void kernel_launch(void* const* d_in, const int* in_sizes, int n_in,
                              void* d_out, int out_size, void* d_ws, size_t ws_size,
                              hipStream_t stream)
{
  (void)in_sizes; (void)n_in; (void)out_size; (void)ws_size;
  const float* x   = (const float*)d_in[0];
  const int*   src = (const int*)  d_in[1];
  const int*   dst = (const int*)  d_in[2];
  const float* W1  = (const float*)d_in[3];
  const float* al1 = (const float*)d_in[4];
  const float* ar1 = (const float*)d_in[5];
  const float* b1  = (const float*)d_in[6];
  const float* W2  = (const float*)d_in[7];
  const float* al2 = (const float*)d_in[8];
  const float* ar2 = (const float*)d_in[9];
  const float* b2  = (const float*)d_in[10];
  const float* W3  = (const float*)d_in[11];
  const float* al3 = (const float*)d_in[12];
  const float* ar3 = (const float*)d_in[13];
  const float* b3  = (const float*)d_in[14];
  float* out = (float*)d_out;

  // Workspace layout (fp32 elements):
  float* f  = (float*)d_ws;                      // [NN,128] projected feats
  float* hA = f  + (size_t)NN * 128;             // layer-1 output
  float* hB = hA + (size_t)NN * 128;             // layer-2 output
  float* ew = hB + (size_t)NN * 128;             // [NE,4] edge scores/weights
  float* el = ew + (size_t)NE * 4;               // [NN,4]
  float* er = el + (size_t)NN * 4;               // [NN,4]
  unsigned* mE = (unsigned*)(er + (size_t)NN * 4); // [NN,4] encoded max
  float* ss = (float*)(mE + (size_t)NN * 4);     // [NN,4] softmax denom

  run_gat_layer(x,  W1, al1, ar1, b1, src, dst, f, hA,  ew, el, er, mE, ss, 4, true,  stream);
  run_gat_layer(hA, W2, al2, ar2, b2, src, dst, f, hB,  ew, el, er, mE, ss, 4, true,  stream);
  run_gat_layer(hB, W3, al3, ar3, b3, src, dst, f, out, ew, el, er, mE, ss, 1, false, stream);
  (void)al3; (void)ar3;
}